// GATClassifier_23716809409340
// MI455X (gfx1250) — compile-verified
//
#include <hip/hip_runtime.h>
#include <math.h>

typedef float v2f __attribute__((ext_vector_type(2)));
typedef float v4f __attribute__((ext_vector_type(4)));
typedef float v8f __attribute__((ext_vector_type(8)));

#define WAVE 32
#define TPB 256

// ---------------------------------------------------------------------------
// C[M,N] = A[M,K] @ B[K,N], row-major f32, via V_WMMA_F32_16X16X4_F32.
// Block = 8 waves, each wave owns one 16-row M-tile and NT 16-col N-tiles.
// B panel (16*NT cols x K) staged transposed in LDS with padded stride
// KP = K+2 (even -> 8B-aligned ds_load_b64; odd (KP/2) 8B lane stride -> no bank
// conflicts). Fragment loads are software-pipelined: all NT B fragments are
// fetched before the WMMA group, and the next A fragment is prefetched so
// its global latency hides behind the current WMMAs.
// Requires N % (16*NT) == 0, K % 4 == 0 (true for all GEMMs here).
// ---------------------------------------------------------------------------
template <int NT>
__global__ void gemm_wmma_lds(const float* __restrict__ A, const float* __restrict__ B,
                              float* __restrict__ C, int M, int N, int K) {
  extern __shared__ float Bt[];            // [16*NT][K+2]
  const int KP = K + 2;
  const int ncols = 16 * NT;
  int tid = threadIdx.x;
  int wid = tid >> 5;
  int lane = tid & (WAVE - 1);
  int ngroup = blockIdx.y;

  // stage Bt[n][k] = B[k][ngroup*ncols + n]; float4 reads coalesced over n
  {
    const int n4s = ncols / 4;
    for (int idx = tid; idx < n4s * K; idx += TPB) {
      int n4 = (idx % n4s) * 4;
      int k  = idx / n4s;
      v4f v = *(const v4f*)(B + (size_t)k * N + (size_t)ngroup * ncols + n4);
#pragma unroll
      for (int j = 0; j < 4; ++j) Bt[(n4 + j) * KP + k] = v[j];
    }
  }
  __syncthreads();

  int mtiles = (M + 15) >> 4;
  int tm = blockIdx.x * 8 + wid;
  if (tm >= mtiles) tm = mtiles - 1;       // clamp: duplicate waves store identical data
  int half = lane >> 4;                    // 0: K pair {0,1}, 1: K pair {2,3}
  int l16  = lane & 15;

  const float* Arow = A + (size_t)(tm * 16 + l16) * K;   // A row m = l16
  const float* Brow = &Bt[l16 * KP + half * 2];          // lane's base in LDS panel
  v8f acc[NT];
#pragma unroll
  for (int t = 0; t < NT; ++t) acc[t] = (v8f){};

  v2f a = *(const v2f*)(Arow + half * 2);  // k = 0 fragment
  for (int k = 0; k < K; k += 4) {
    v2f bf[NT];
#pragma unroll
    for (int t = 0; t < NT; ++t)
      bf[t] = *(const v2f*)(Brow + (size_t)t * 16 * KP + k);
    v2f an = a;
    if (k + 4 < K) an = *(const v2f*)(Arow + k + 4 + half * 2);  // prefetch next A
#pragma unroll
    for (int t = 0; t < NT; ++t)
      acc[t] = __builtin_amdgcn_wmma_f32_16x16x4_f32(false, a, false, bf[t],
                                                     (short)0, acc[t], false, false);
    a = an;
  }

  // D layout: VGPR v, lane L -> row (v + 8*half), col l16
#pragma unroll
  for (int t = 0; t < NT; ++t) {
    size_t cbase = (size_t)(tm * 16 + half * 8) * N +
                   (size_t)ngroup * ncols + (size_t)t * 16 + l16;
#pragma unroll
    for (int v = 0; v < 8; ++v) C[cbase + (size_t)v * N] = acc[t][v];
  }
}

// asrc[n,h] = <h[n,h,:], a_src[h,:]> ; adst likewise
__global__ void attn_scores(const float* __restrict__ h, const float* __restrict__ a_src,
                            const float* __restrict__ a_dst, float* __restrict__ asrc,
                            float* __restrict__ adst, int n, int H, int D) {
  int t = blockIdx.x * blockDim.x + threadIdx.x;
  if (t >= n * H) return;
  int node = t / H, hd = t - node * H;
  const float* hv = h + (size_t)node * H * D + (size_t)hd * D;
  const float* as = a_src + hd * D;
  const float* ad = a_dst + hd * D;
  float s1 = 0.f, s2 = 0.f;
  for (int d = 0; d < D; ++d) { float v = hv[d]; s1 += v * as[d]; s2 += v * ad[d]; }
  asrc[t] = s1;
  adst[t] = s2;
}

__global__ void fillf(float* __restrict__ p, float v, int n) {
  int t = blockIdx.x * blockDim.x + threadIdx.x;
  if (t < n) p[t] = v;
}

__device__ __forceinline__ void atomicMaxF(float* addr, float val) {
  // order-preserving int encoding; works with -inf init and mixed signs
  if (val >= 0.f) atomicMax((int*)addr, __float_as_int(val));
  else            atomicMin((unsigned int*)addr, __float_as_uint(val));
}

__device__ __forceinline__ void edge_endpoints(const int* __restrict__ ei, int E, int e,
                                               int& src, int& dst) {
  if (e < E) { src = ei[e]; dst = ei[E + e]; }
  else       { src = dst = e - E; }          // appended self-loops
}

// pass 1: m[dst,h] = segment_max(leaky_relu(asrc[src]+adst[dst]))
__global__ void edge_max(const int* __restrict__ ei, int E, int n, int H,
                         const float* __restrict__ asrc, const float* __restrict__ adst,
                         float* __restrict__ m) {
  int t = blockIdx.x * blockDim.x + threadIdx.x;
  if (t >= (E + n) * H) return;
  int e = t / H, hd = t - e * H;
  int src, dst; edge_endpoints(ei, E, e, src, dst);
  float x = asrc[src * H + hd] + adst[dst * H + hd];
  float ev = x >= 0.f ? x : 0.2f * x;
  atomicMaxF(&m[dst * H + hd], ev);
}

// pass 2: p = exp(e - m[dst]); s[dst,h] += p; store p for pass 3
__global__ void edge_expsum(const int* __restrict__ ei, int E, int n, int H,
                            const float* __restrict__ asrc, const float* __restrict__ adst,
                            const float* __restrict__ m, float* __restrict__ p,
                            float* __restrict__ s) {
  int t = blockIdx.x * blockDim.x + threadIdx.x;
  if (t >= (E + n) * H) return;
  int e = t / H, hd = t - e * H;
  int src, dst; edge_endpoints(ei, E, e, src, dst);
  float x = asrc[src * H + hd] + adst[dst * H + hd];
  float ev = x >= 0.f ? x : 0.2f * x;
  float pe = expf(ev - m[dst * H + hd]);
  p[t] = pe;
  atomicAdd(&s[dst * H + hd], pe);
}

// pass 3: out[dst,h,:] += alpha * h[src,h,:], one wave per edge.
// alpha per head computed once by lanes 0..H-1, broadcast via shuffle.
__global__ void edge_scatter(const int* __restrict__ ei, int E, int n, int H, int D,
                             const float* __restrict__ p, const float* __restrict__ s,
                             const float* __restrict__ h, float* __restrict__ out) {
  int gid  = blockIdx.x * blockDim.x + threadIdx.x;
  int e    = gid / WAVE;
  int lane = threadIdx.x & (WAVE - 1);
  if (e >= E + n) return;                   // wave-uniform
  int src, dst; edge_endpoints(ei, E, e, src, dst);
  float alpha_l = 0.f;
  if (lane < H)
    alpha_l = p[(size_t)e * H + lane] / (s[dst * H + lane] + 1e-16f);
  int HD = H * D;
  const float* hs = h + (size_t)src * HD;
  float* od = out + (size_t)dst * HD;
  for (int idx = lane; idx < HD; idx += WAVE) {
    float alpha = __shfl(alpha_l, idx / D);
    atomicAdd(&od[idx], hs[idx] * alpha);
  }
}

__global__ void bias_elu(float* __restrict__ out, const float* __restrict__ b,
                         long long total, int HD) {
  long long t = (long long)blockIdx.x * blockDim.x + threadIdx.x;
  if (t >= total) return;
  float v = out[t] + b[t % HD];
  out[t] = v > 0.f ? v : expf(v) - 1.f;     // ELU, alpha=1
}

__global__ void bias_add(float* __restrict__ out, const float* __restrict__ b,
                         long long total, int HD) {
  long long t = (long long)blockIdx.x * blockDim.x + threadIdx.x;
  if (t >= total) return;
  out[t] += b[t % HD];
}

__global__ void pool_sum(const float* __restrict__ h, const int* __restrict__ batch,
                         float* __restrict__ pooled, float* __restrict__ cnt, int n, int D) {
  int t = blockIdx.x * blockDim.x + threadIdx.x;
  if (t >= n * D) return;
  int node = t / D, d = t - node * D;
  int g = batch[node];
  atomicAdd(&pooled[g * D + d], h[t]);
  if (d == 0) atomicAdd(&cnt[g], 1.0f);
}

__global__ void fc_out(const float* __restrict__ pooled, const float* __restrict__ cnt,
                       const float* __restrict__ W, const float* __restrict__ b,
                       float* __restrict__ out, int G, int D, int C) {
  int t = blockIdx.x * blockDim.x + threadIdx.x;
  if (t >= G * C) return;
  int g = t / C, c = t - g * C;
  float inv = 1.0f / fmaxf(cnt[g], 1.0f);
  float acc = b[c];
  for (int d = 0; d < D; ++d) acc += pooled[g * D + d] * inv * W[d * C + c];
  out[t] = acc;
}

extern "C" void kernel_launch(void* const* d_in, const int* in_sizes, int n_in,
                              void* d_out, int out_size, void* d_ws, size_t ws_size,
                              hipStream_t stream) {
  const float* x      = (const float*)d_in[0];
  const int*   ei     = (const int*)  d_in[1];
  const int*   batch  = (const int*)  d_in[2];
  const float* W1     = (const float*)d_in[3];
  const float* a_src1 = (const float*)d_in[4];
  const float* a_dst1 = (const float*)d_in[5];
  const float* b1     = (const float*)d_in[6];
  const float* W2     = (const float*)d_in[7];
  const float* a_src2 = (const float*)d_in[8];
  const float* a_dst2 = (const float*)d_in[9];
  const float* b2     = (const float*)d_in[10];
  const float* fcW    = (const float*)d_in[11];
  const float* fcb    = (const float*)d_in[12];

  const int N = in_sizes[2];            // batch has one entry per node
  const int E = in_sizes[1] / 2;
  const int IN_DIM = 128, HID = 64, HEADS = 4, NC = 16, G = 64;
  const int HD1 = HEADS * HID;          // 256

  // ---- workspace layout (f32, 64-element aligned) ----
  float* w = (float*)d_ws;
  size_t off = 0;
  auto alloc = [&](size_t nf) { float* q = w + off; off += (nf + 63) & ~(size_t)63; return q; };
  float* h1    = alloc((size_t)N * HD1);
  float* out1  = alloc((size_t)N * HD1);
  float* h2    = alloc((size_t)N * HID);
  float* out2  = alloc((size_t)N * HID);
  float* asrc1 = alloc((size_t)N * HEADS);
  float* adst1 = alloc((size_t)N * HEADS);
  float* m1    = alloc((size_t)N * HEADS);
  float* s1    = alloc((size_t)N * HEADS);
  float* asrc2 = alloc(N);
  float* adst2 = alloc(N);
  float* m2    = alloc(N);
  float* s2    = alloc(N);
  float* pbuf  = alloc((size_t)(E + N) * HEADS);   // reused for layer 2 (H=1)
  float* pooled= alloc((size_t)G * HID);
  float* cnt   = alloc(G);

  // zero the accumulators every call (graph-capturable memsets)
  hipMemsetAsync(out1,   0, (size_t)N * HD1 * sizeof(float), stream);
  hipMemsetAsync(out2,   0, (size_t)N * HID * sizeof(float), stream);
  hipMemsetAsync(s1,     0, (size_t)N * HEADS * sizeof(float), stream);
  hipMemsetAsync(s2,     0, (size_t)N * sizeof(float), stream);
  hipMemsetAsync(pooled, 0, (size_t)G * HID * sizeof(float), stream);
  hipMemsetAsync(cnt,    0, (size_t)G * sizeof(float), stream);

  auto nb = [&](long long t) { return (int)((t + TPB - 1) / TPB); };
  int mtiles  = (N + 15) / 16;
  int mblocks = (mtiles + 7) / 8;

  // ---------------- Layer 1 (H=4, D=64, concat) ----------------
  {
    dim3 grid(mblocks, HD1 / (16 * 4));                 // 4 N-tiles per wave
    size_t smem = (size_t)64 * (IN_DIM + 2) * sizeof(float);   // ~33 KB
    gemm_wmma_lds<4><<<grid, TPB, smem, stream>>>(x, W1, h1, N, HD1, IN_DIM);
  }
  attn_scores<<<nb((long long)N * HEADS), TPB, 0, stream>>>(h1, a_src1, a_dst1, asrc1, adst1, N, HEADS, HID);
  fillf<<<nb((long long)N * HEADS), TPB, 0, stream>>>(m1, -INFINITY, N * HEADS);
  edge_max   <<<nb((long long)(E + N) * HEADS), TPB, 0, stream>>>(ei, E, N, HEADS, asrc1, adst1, m1);
  edge_expsum<<<nb((long long)(E + N) * HEADS), TPB, 0, stream>>>(ei, E, N, HEADS, asrc1, adst1, m1, pbuf, s1);
  edge_scatter<<<nb((long long)(E + N) * WAVE), TPB, 0, stream>>>(ei, E, N, HEADS, HID, pbuf, s1, h1, out1);
  bias_elu<<<nb((long long)N * HD1), TPB, 0, stream>>>(out1, b1, (long long)N * HD1, HD1);

  // ---------------- Layer 2 (H=1, D=64, mean == identity) ----------------
  {
    dim3 grid(mblocks, HID / (16 * 2));                 // 2 N-tiles per wave
    size_t smem = (size_t)32 * (HD1 + 2) * sizeof(float);      // ~33 KB
    gemm_wmma_lds<2><<<grid, TPB, smem, stream>>>(out1, W2, h2, N, HID, HD1);
  }
  attn_scores<<<nb((long long)N), TPB, 0, stream>>>(h2, a_src2, a_dst2, asrc2, adst2, N, 1, HID);
  fillf<<<nb((long long)N), TPB, 0, stream>>>(m2, -INFINITY, N);
  edge_max   <<<nb((long long)(E + N)), TPB, 0, stream>>>(ei, E, N, 1, asrc2, adst2, m2);
  edge_expsum<<<nb((long long)(E + N)), TPB, 0, stream>>>(ei, E, N, 1, asrc2, adst2, m2, pbuf, s2);
  edge_scatter<<<nb((long long)(E + N) * WAVE), TPB, 0, stream>>>(ei, E, N, 1, HID, pbuf, s2, h2, out2);
  bias_add<<<nb((long long)N * HID), TPB, 0, stream>>>(out2, b2, (long long)N * HID, HID);

  // ---------------- global mean pool + FC ----------------
  pool_sum<<<nb((long long)N * HID), TPB, 0, stream>>>(out2, batch, pooled, cnt, N, HID);
  fc_out<<<nb((long long)G * NC), TPB, 0, stream>>>(pooled, cnt, fcW, fcb, (float*)d_out, G, HID, NC);
}